// OriginalGlassBlur_14757507629596
// MI455X (gfx1250) — compile-verified
//
#include <hip/hip_runtime.h>

// ---------------------------------------------------------------------------
// Glass blur for MI455X (gfx1250, wave32, WMMA).
//   blur  = two dense 224x224 matmuls per channel (banded reflect-pad blur
//           matrix), executed with V_WMMA_F32_16X16X4_F32 (fp32 end to end).
//   swaps = exact-order-preserving wavefront schedule (t = 4*i1 + i2) on a
//           per-channel 196KB LDS image (fits CDNA5's 320KB/WGP LDS).
//   data  = global_load_async_to_lds_b128 (ASYNCcnt) + global_prefetch_b8.
// ---------------------------------------------------------------------------

typedef __attribute__((ext_vector_type(2))) float v2f;
typedef __attribute__((ext_vector_type(8))) float v8f;

constexpr int IH    = 224;
constexpr int IW    = 224;
constexpr int PLANE = IH * IW;          // 50176 floats = 200704 bytes
constexpr int GMD   = 2;                // MD
constexpr int GITER = 3;                // ITER
constexpr int RD    = IH - GMD;         // 222 : rands spatial dim
constexpr int LH    = IH - GMD - GMD + 0; // hs/ws length = arange(222,2,-1) -> 220
// LH really is 220:
static_assert(RD - GMD == 220, "");

// Normalized 5-tap Gaussian, sigma = 1 (exact to fp32):
// w = exp(-{4,1,0,1,4}/2) / sum
constexpr float G0 = 0.054488684549644346f;  // exp(-2)/Z
constexpr float G1 = 0.24420134200323332f;   // exp(-0.5)/Z
constexpr float G2 = 0.40261994689424467f;   // 1/Z

// ---------------------------------------------------------------------------
// Build banded blur matrix B (and its transpose) with 'reflect' boundary:
//   out[i] = sum_k g[k] * in[reflect(i+k-2)]   =>   Out = B * In
// ---------------------------------------------------------------------------
__global__ void build_blur_mats(float* __restrict__ B, float* __restrict__ BT) {
    int idx = blockIdx.x * blockDim.x + threadIdx.x;
    if (idx >= IH * IH) return;
    int i = idx / IH;
    int j = idx % IH;
    const float g[5] = {G0, G1, G2, G1, G0};
    float v = 0.0f;
#pragma unroll
    for (int k = 0; k < 5; ++k) {
        int s = i + k - 2;
        if (s < 0) s = -s;                       // reflect (no edge repeat)
        if (s > IH - 1) s = 2 * (IH - 1) - s;
        if (s == j) v += g[k];
    }
    B [i * IH + j] = v;
    BT[j * IH + i] = v;
}

// ---------------------------------------------------------------------------
// Pack (dx,dy) in [-2,1] into one byte per swap: ((dy+2)<<4)|(dx+2).
// rands layout: (ITER, 222, 222, 2) int32; we use the [:220,:220] slice.
// ---------------------------------------------------------------------------
__global__ void pack_rands(const int* __restrict__ rands,
                           unsigned char* __restrict__ packed) {
    int idx = blockIdx.x * blockDim.x + threadIdx.x;
    if (idx >= GITER * 220 * 220) return;
    int it  = idx / (220 * 220);
    int rem = idx % (220 * 220);
    int i1  = rem / 220;
    int i2  = rem % 220;
    int src = ((it * RD + i1) * RD + i2) * 2;
    int dx  = rands[src + 0];
    int dy  = rands[src + 1];
    packed[idx] = (unsigned char)(((dy + 2) << 4) | ((dx + 2) & 0xF));
}

// ---------------------------------------------------------------------------
// 224x224x224 fp32 matmul, one wave per 16x16 output tile, K-step 4 via
// V_WMMA_F32_16X16X4_F32. aCh/bCh/oCh = per-channel element strides (0 if
// the operand is shared across channels). clip=1 -> clamp to [0,1].
//
// Fragment layouts (ISA 7.12.2, wave32):
//   A 16x4 f32 : lane L holds A[L&15][kb], A[L&15][kb+1], kb = (L>>4)*2
//   B 4x16 f32 : lane L holds B[kb][L&15],  B[kb+1][L&15]
//   C/D 16x16  : vgpr r, lane L -> D[r + (L>>4)*8][L&15]
// ---------------------------------------------------------------------------
__global__ __launch_bounds__(32)
void wmma_mm224(const float* __restrict__ A, long aCh,
                const float* __restrict__ Bm, long bCh,
                float* __restrict__ O, long oCh, int clip) {
    const int tile = blockIdx.x;            // 0..195
    const int ch   = blockIdx.y;            // 0..2
    const int m0   = (tile / 14) * 16;
    const int n0   = (tile % 14) * 16;
    const int lane = threadIdx.x;
    const int half = lane >> 4;             // 0 or 1
    const int l15  = lane & 15;
    const int kb   = half * 2;

    const float* Ap = A  + (long)ch * aCh;
    const float* Bp = Bm + (long)ch * bCh;
    float*       Op = O  + (long)ch * oCh;

    v8f c = {};
    for (int k0 = 0; k0 < 224; k0 += 4) {
        v2f a, b;
        const float* ap = Ap + (m0 + l15) * IW + k0 + kb;
        a.x = ap[0];
        a.y = ap[1];
        const float* bp = Bp + (k0 + kb) * IW + n0 + l15;
        b.x = bp[0];
        b.y = bp[IW];
        // (neg_a, A, neg_b, B, c_mod, C, reuse_a, reuse_b)
        c = __builtin_amdgcn_wmma_f32_16x16x4_f32(
                false, a, false, b, (short)0, c, false, false);
    }
#pragma unroll
    for (int r = 0; r < 8; ++r) {
        float v = c[r];
        if (clip) v = fminf(fmaxf(v, 0.0f), 1.0f);
        Op[(m0 + r + half * 8) * IW + n0 + l15] = v;
    }
}

// ---------------------------------------------------------------------------
// Exact-order glass swaps. One block per channel; 196KB channel image lives
// in LDS. Wavefront time t = 4*i1 + i2: every dependency of a swap has a
// strictly smaller t (offsets span rows [h-2,h+1], cols [w-2,w+1]; equal-t
// swaps differ by >=4 columns -> disjoint), so equal-t swaps run in parallel
// (<=55 active lanes) separated by workgroup barriers.
// ---------------------------------------------------------------------------
__global__ __launch_bounds__(64)
void glass_swap(const float* __restrict__ src, float* __restrict__ dst,
                const unsigned char* __restrict__ packed) {
    extern __shared__ float sm[];           // PLANE floats (dynamic LDS)
    const int c   = blockIdx.x;
    const int tid = threadIdx.x;
    const float* s = src + c * PLANE;

    // Async DMA of the channel into LDS: 12544 b128 transfers, ASYNCcnt path.
    for (int i = tid; i < PLANE / 4; i += 64) {
        int laddr = i * 16;                 // LDS byte address (dyn LDS base 0)
        int goff  = i * 16;                 // byte offset from SGPR base (GVS)
        asm volatile("global_load_async_to_lds_b128 %0, %1, %2"
                     :: "v"(laddr), "v"(goff), "s"(s) : "memory");
    }
    asm volatile("s_wait_asynccnt 0" ::: "memory");
    __syncthreads();

    for (int it = 0; it < GITER; ++it) {
        const unsigned char* pit = packed + it * 220 * 220;
        // Warm L2/WGP$ for this iteration's swap table.
        for (int i = tid * 128; i < 220 * 220; i += 64 * 128)
            __builtin_prefetch(pit + i, 0, 1);

        for (int t = 0; t <= 4 * 219 + 219; ++t) {
            int i1min = t - 219;
            i1min = (i1min > 0) ? ((i1min + 3) >> 2) : 0;   // ceil((t-219)/4)
            int i1max = t >> 2;
            if (i1max > 219) i1max = 219;
            int i1 = i1min + tid;
            if (i1 <= i1max) {
                int i2 = t - 4 * i1;                        // in [0,219]
                unsigned int r = pit[i1 * 220 + i2];
                int dy = (int)(r >> 4) - 2;
                int dx = (int)(r & 15) - 2;
                int h  = 222 - i1;                          // hs[i1]
                int w  = 222 - i2;                          // ws[i2]
                int p  = h * IW + w;
                int q  = (h + dy) * IW + (w + dx);
                float va = sm[p];
                float vb = sm[q];
                sm[p] = vb;
                sm[q] = va;
            }
            __syncthreads();
        }
    }

    float* d = dst + c * PLANE;
    for (int i = tid; i < PLANE; i += 64) d[i] = sm[i];
}

// ---------------------------------------------------------------------------
// Workspace layout (bytes), all 16B aligned. Total ~1.75 MB.
// ---------------------------------------------------------------------------
extern "C" void kernel_launch(void* const* d_in, const int* in_sizes, int n_in,
                              void* d_out, int out_size, void* d_ws, size_t ws_size,
                              hipStream_t stream) {
    const float* img   = (const float*)d_in[0];   // (3,224,224) f32
    const int*   rands = (const int*)d_in[1];     // (3,222,222,2) i32
    char* ws = (char*)d_ws;

    float*         Bmat   = (float*)(ws + 0);        // 200704 B
    float*         BTmat  = (float*)(ws + 200704);   // 200704 B
    float*         tmp0   = (float*)(ws + 401408);   // 602112 B
    float*         tmp1   = (float*)(ws + 1003520);  // 602112 B
    unsigned char* packed = (unsigned char*)(ws + 1605632); // 145200 B
    float*         out    = (float*)d_out;

    build_blur_mats<<<(IH * IH + 255) / 256, 256, 0, stream>>>(Bmat, BTmat);
    pack_rands<<<(GITER * 220 * 220 + 255) / 256, 256, 0, stream>>>(rands, packed);

    // Blur pass 1: vertical (Y = B * X), then horizontal (Z = Y * B^T).
    wmma_mm224<<<dim3(196, 3), 32, 0, stream>>>(Bmat, 0,     img,  PLANE, tmp0, PLANE, 0);
    wmma_mm224<<<dim3(196, 3), 32, 0, stream>>>(tmp0, PLANE, BTmat, 0,    tmp1, PLANE, 0);

    // Order-exact pixel swap chain, channel-parallel across WGPs.
    glass_swap<<<3, 64, PLANE * sizeof(float), stream>>>(tmp1, tmp0, packed);

    // Blur pass 2 + clip to [0,1].
    wmma_mm224<<<dim3(196, 3), 32, 0, stream>>>(Bmat, 0,     tmp0, PLANE, tmp1, PLANE, 0);
    wmma_mm224<<<dim3(196, 3), 32, 0, stream>>>(tmp1, PLANE, BTmat, 0,    out,  PLANE, 1);
}